// ImprovedMambaBlock_59828894433548
// MI455X (gfx1250) — compile-verified
//
#include <hip/hip_runtime.h>
#include <math.h>

#define D_MODEL  1024
#define D_STATE  16
#define D_INNER  2048
#define BB       4
#define LL       1024
#define M_TOK    (BB*LL)        // 4096 tokens

typedef float v2f __attribute__((ext_vector_type(2)));
typedef float v8f __attribute__((ext_vector_type(8)));

__device__ __forceinline__ float wave_sum32(float v) {
#pragma unroll
    for (int off = 16; off >= 1; off >>= 1)
        v += __shfl_xor(v, off, 32);
    return v;
}

// gfx1250 async memory->LDS copy, 16B per lane, tracked by ASYNCcnt
__device__ __forceinline__ void async_ld16(unsigned lds_byte, const float* gp) {
    asm volatile("global_load_async_to_lds_b128 %0, %1, off"
                 :: "v"(lds_byte), "v"(gp) : "memory");
}

// ---------------------------------------------------------------------------
// 1) LayerNorm: one wave per row of 1024, row kept in registers (32/lane)
// ---------------------------------------------------------------------------
__global__ __launch_bounds__(256) void ln_kernel(const float* __restrict__ x,
                                                 const float* __restrict__ w,
                                                 const float* __restrict__ b,
                                                 float* __restrict__ out) {
    int wave = threadIdx.x >> 5;
    int lane = threadIdx.x & 31;
    int row  = blockIdx.x * 8 + wave;          // grid = 512 -> 4096 rows
    const float* xr = x + (size_t)row * D_MODEL;
    float v[32];
    float s = 0.f, sq = 0.f;
#pragma unroll
    for (int i = 0; i < 32; ++i) {
        float t = xr[lane + i * 32];
        v[i] = t; s += t; sq += t * t;
    }
    s  = wave_sum32(s);
    sq = wave_sum32(sq);
    float mu  = s * (1.0f / D_MODEL);
    float var = sq * (1.0f / D_MODEL) - mu * mu;
    float rs  = rsqrtf(var + 1e-5f);
    float* orow = out + (size_t)row * D_MODEL;
#pragma unroll
    for (int i = 0; i < 32; ++i) {
        int c = lane + i * 32;
        orow[c] = (v[i] - mu) * rs * w[c] + b[c];
    }
}

// ---------------------------------------------------------------------------
// 2) fp32 WMMA GEMM:  C[M,N] = A[M,K] * W[N,K]^T  (+ epilogue)
//    EPI 0: none   EPI 1: softplus(acc + bias[n])   EPI 2: acc + res[m*ldc+n]
//    Block tile 128x128, 8 waves (4x2), wave tile 32x64 (2x4 of 16x16).
//    K staged in steps of 16 through double-buffered LDS filled with
//    global_load_async_to_lds_b128 (ASYNCcnt pipeline).
// ---------------------------------------------------------------------------
template <int EPI>
__global__ __launch_bounds__(256) void gemm_kernel(const float* __restrict__ A, int lda,
                                                   const float* __restrict__ W, int ldw,
                                                   float* __restrict__ C, int ldc,
                                                   const float* __restrict__ aux,
                                                   int K) {
    // 16 K-cols + 4 pad -> conflict-free fragment reads; 40KB total
    __shared__ float As[2][128][20];
    __shared__ float Bs[2][128][20];   // Bs[n][k] = W[n][k] (straight copy)

    int tid    = threadIdx.x;
    int lane   = tid & 31;
    int waveId = tid >> 5;
    int wm     = waveId >> 1;       // 0..3  (rows of 32)
    int wn     = waveId & 1;        // 0..1  (cols of 64)
    int lhalf  = lane >> 4;         // K-pair select per ISA 16x4 layout
    int l15    = lane & 15;
    int m0     = blockIdx.y * 128;
    int n0     = blockIdx.x * 128;

    v8f acc[2][4] = {};

    // each thread owns two 16B chunks of A tile and two of B tile
    int c0 = tid, c1 = tid + 256;            // chunk ids, 512 chunks per tile
    int ar0 = c0 >> 2, ac0 = (c0 & 3) << 2;
    int ar1 = c1 >> 2, ac1 = (c1 & 3) << 2;

    const float* aptr0 = A + (size_t)(m0 + ar0) * lda + ac0;
    const float* aptr1 = A + (size_t)(m0 + ar1) * lda + ac1;
    const float* bptr0 = W + (size_t)(n0 + ar0) * ldw + ac0;
    const float* bptr1 = W + (size_t)(n0 + ar1) * ldw + ac1;

    unsigned lA0[2], lA1[2], lB0[2], lB1[2];
#pragma unroll
    for (int bf = 0; bf < 2; ++bf) {
        lA0[bf] = (unsigned)(uintptr_t)&As[bf][ar0][ac0];
        lA1[bf] = (unsigned)(uintptr_t)&As[bf][ar1][ac1];
        lB0[bf] = (unsigned)(uintptr_t)&Bs[bf][ar0][ac0];
        lB1[bf] = (unsigned)(uintptr_t)&Bs[bf][ar1][ac1];
    }

    const int T = K >> 4;
    // prologue: tile 0 -> buffer 0 (4 async ops per wave in flight)
    async_ld16(lA0[0], aptr0);
    async_ld16(lA1[0], aptr1);
    async_ld16(lB0[0], bptr0);
    async_ld16(lB1[0], bptr1);

    for (int t = 0; t < T; ++t) {
        int buf = t & 1;
        if (t + 1 < T) {                      // prefetch tile t+1 into other buf
            int k1 = (t + 1) << 4;
            int nb = buf ^ 1;
            async_ld16(lA0[nb], aptr0 + k1);
            async_ld16(lA1[nb], aptr1 + k1);
            async_ld16(lB0[nb], bptr0 + k1);
            async_ld16(lB1[nb], bptr1 + k1);
            // in-order completion: <=4 outstanding => tile t's 4 are done
            asm volatile("s_wait_asynccnt 0x4" ::: "memory");
        } else {
            asm volatile("s_wait_asynccnt 0x0" ::: "memory");
        }
        __syncthreads();                      // tile t visible to all waves

#pragma unroll
        for (int kk = 0; kk < 16; kk += 4) {
            v2f afr[2], bfr[4];
#pragma unroll
            for (int mi = 0; mi < 2; ++mi) {
                int r = wm * 32 + mi * 16 + l15;
                afr[mi][0] = As[buf][r][kk + 2 * lhalf];
                afr[mi][1] = As[buf][r][kk + 2 * lhalf + 1];
            }
#pragma unroll
            for (int ni = 0; ni < 4; ++ni) {
                int r = wn * 64 + ni * 16 + l15;
                bfr[ni][0] = Bs[buf][r][kk + 2 * lhalf];
                bfr[ni][1] = Bs[buf][r][kk + 2 * lhalf + 1];
            }
#pragma unroll
            for (int mi = 0; mi < 2; ++mi)
#pragma unroll
                for (int ni = 0; ni < 4; ++ni)
                    acc[mi][ni] = __builtin_amdgcn_wmma_f32_16x16x4_f32(
                        false, afr[mi], false, bfr[ni],
                        (short)0, acc[mi][ni], false, false);
        }
        __syncthreads();                      // done reading buf before refill
    }

#pragma unroll
    for (int mi = 0; mi < 2; ++mi) {
#pragma unroll
        for (int ni = 0; ni < 4; ++ni) {
            int col = n0 + wn * 64 + ni * 16 + l15;
#pragma unroll
            for (int j = 0; j < 8; ++j) {
                int row = m0 + wm * 32 + mi * 16 + j + 8 * lhalf;
                float val = acc[mi][ni][j];
                if (EPI == 1) {                       // softplus(acc + bias)
                    float z = val + aux[col];
                    val = (z > 20.f) ? z : log1pf(expf(z));
                } else if (EPI == 2) {                // residual add
                    val += aux[(size_t)row * ldc + col];
                }
                C[(size_t)row * ldc + col] = val;
            }
        }
    }
}

// ---------------------------------------------------------------------------
// 3) causal depthwise conv (K=4) + SiLU; reads x_branch = xg[:, 0:2048]
// ---------------------------------------------------------------------------
__global__ __launch_bounds__(256) void conv_silu_kernel(const float* __restrict__ xg,
                                                        const float* __restrict__ cw,
                                                        const float* __restrict__ cb,
                                                        float* __restrict__ xc) {
    int idx = blockIdx.x * 256 + threadIdx.x;  // m*2048 + c
    int c   = idx & (D_INNER - 1);
    int m   = idx >> 11;
    int l   = m & (LL - 1);
    int b0  = m - l;                            // b*L
    float acc = cb[c];
#pragma unroll
    for (int k = 0; k < 4; ++k) {
        int ls = l + k - 3;
        if (ls >= 0)
            acc += xg[(size_t)(b0 + ls) * 4096 + c] * cw[c * 4 + k];
    }
    xc[idx] = acc / (1.f + expf(-acc));          // silu
}

// ---------------------------------------------------------------------------
// 4) B/C projections (N=16 each): block per token, token row cached in LDS
// ---------------------------------------------------------------------------
__global__ __launch_bounds__(256) void bc_proj_kernel(const float* __restrict__ xc,
                                                      const float* __restrict__ Bw,
                                                      const float* __restrict__ Cw,
                                                      float* __restrict__ Bp,
                                                      float* __restrict__ Cp) {
    __shared__ float sh[D_INNER];
    int m   = blockIdx.x;
    int tid = threadIdx.x;
#pragma unroll
    for (int i = 0; i < 8; ++i)
        sh[tid + i * 256] = xc[(size_t)m * D_INNER + tid + i * 256];
    __syncthreads();
    int wave = tid >> 5, lane = tid & 31;
#pragma unroll
    for (int q = 0; q < 4; ++q) {
        int o = wave * 4 + q;                    // 0..31 (16 Bp + 16 Cp)
        const float* Wr = (o < 16) ? (Bw + (size_t)o * D_INNER)
                                   : (Cw + (size_t)(o - 16) * D_INNER);
        float s = 0.f;
        for (int ci = lane; ci < D_INNER; ci += 32)
            s += sh[ci] * Wr[ci];
        s = wave_sum32(s);
        if (lane == 0) {
            if (o < 16) Bp[m * 16 + o] = s;
            else        Cp[m * 16 + (o - 16)] = s;
        }
    }
}

// ---------------------------------------------------------------------------
// 5) selective scan: lane = (n, d-parity), 16 n-lanes per channel,
//    131072 independent length-1024 recurrences
// ---------------------------------------------------------------------------
__global__ __launch_bounds__(256) void ssm_scan_kernel(const float* __restrict__ dt,
                                                       const float* __restrict__ Bp,
                                                       const float* __restrict__ Cp,
                                                       const float* __restrict__ xc,
                                                       const float* __restrict__ A_log,
                                                       const float* __restrict__ Dp,
                                                       float* __restrict__ yss) {
    int gtid = blockIdx.x * 256 + threadIdx.x;
    int lane = gtid & 31;
    int half = lane >> 4;
    int n    = lane & 15;
    int ch   = (gtid >> 5) * 2 + half;           // (b,d) channel, 0..8191
    int b    = ch >> 11;
    int d    = ch & (D_INNER - 1);

    float a  = -expf(A_log[d * 16 + n]);
    float Dv = Dp[d];
    float h  = 0.f;
    size_t tok = (size_t)b * LL;
    for (int l = 0; l < LL; ++l, ++tok) {
        float dtv = dt[tok * D_INNER + d];
        float xv  = xc[tok * D_INNER + d];
        float bpv = Bp[tok * 16 + n];
        float cpv = Cp[tok * 16 + n];
        float dA  = __expf(dtv * a);             // v_exp_f32
        h = dA * h + dtv * bpv * xv;
        float y = h * cpv;
        y += __shfl_xor(y, 1, 32);
        y += __shfl_xor(y, 2, 32);
        y += __shfl_xor(y, 4, 32);
        y += __shfl_xor(y, 8, 32);               // sum over 16 n-lanes
        if (n == 0)
            yss[tok * D_INNER + d] = y + Dv * xv;
    }
}

// ---------------------------------------------------------------------------
// 6) gate (in place into gate half of xg): g = yss * silu(gate)
// ---------------------------------------------------------------------------
__global__ __launch_bounds__(256) void gate_kernel(const float* __restrict__ yss,
                                                   float* __restrict__ xg) {
    int idx = blockIdx.x * 256 + threadIdx.x;    // m*2048 + e
    int e = idx & (D_INNER - 1);
    int m = idx >> 11;
    float* gp = xg + (size_t)m * 4096 + 2048 + e;
    float g  = *gp;
    float sg = g / (1.f + expf(-g));
    *gp = yss[idx] * sg;
}

// ---------------------------------------------------------------------------
extern "C" void kernel_launch(void* const* d_in, const int* in_sizes, int n_in,
                              void* d_out, int out_size, void* d_ws, size_t ws_size,
                              hipStream_t stream) {
    const float* x         = (const float*)d_in[0];
    const float* norm_w    = (const float*)d_in[1];
    const float* norm_b    = (const float*)d_in[2];
    const float* in_proj_w = (const float*)d_in[3];
    const float* conv_w    = (const float*)d_in[4];
    const float* conv_b    = (const float*)d_in[5];
    const float* dt_w      = (const float*)d_in[6];
    const float* dt_b      = (const float*)d_in[7];
    const float* Bp_w      = (const float*)d_in[8];
    const float* Cp_w      = (const float*)d_in[9];
    const float* A_log     = (const float*)d_in[10];
    const float* D_param   = (const float*)d_in[11];
    const float* out_pw    = (const float*)d_in[12];
    float* out = (float*)d_out;

    float* ws  = (float*)d_ws;
    float* xn  = ws;                                  // 4096*1024
    float* xg  = xn  + (size_t)M_TOK * D_MODEL;       // 4096*4096
    float* xc  = xg  + (size_t)M_TOK * 2 * D_INNER;   // 4096*2048
    float* dtb = xc  + (size_t)M_TOK * D_INNER;       // 4096*2048
    float* Bp  = dtb + (size_t)M_TOK * D_INNER;       // 4096*16
    float* Cp  = Bp  + (size_t)M_TOK * D_STATE;       // 4096*16
    float* yss = Cp  + (size_t)M_TOK * D_STATE;       // 4096*2048

    // 1) layernorm
    ln_kernel<<<M_TOK / 8, 256, 0, stream>>>(x, norm_w, norm_b, xn);

    // 2) in_proj: [4096,1024] x [1024->4096]
    gemm_kernel<0><<<dim3(4096 / 128, M_TOK / 128), 256, 0, stream>>>(
        xn, D_MODEL, in_proj_w, D_MODEL, xg, 2 * D_INNER, nullptr, D_MODEL);

    // 3) causal conv + silu
    conv_silu_kernel<<<(M_TOK * D_INNER) / 256, 256, 0, stream>>>(xg, conv_w, conv_b, xc);

    // 4) dt = softplus(xc @ dt_w^T + dt_b)
    gemm_kernel<1><<<dim3(D_INNER / 128, M_TOK / 128), 256, 0, stream>>>(
        xc, D_INNER, dt_w, D_INNER, dtb, D_INNER, dt_b, D_INNER);

    // 5) B/C projections
    bc_proj_kernel<<<M_TOK, 256, 0, stream>>>(xc, Bp_w, Cp_w, Bp, Cp);

    // 6) selective scan (+ D*x skip)
    ssm_scan_kernel<<<(BB * D_INNER * D_STATE) / 256, 256, 0, stream>>>(
        dtb, Bp, Cp, xc, A_log, D_param, yss);

    // 7) gating, in place into xg's gate half
    gate_kernel<<<(M_TOK * D_INNER) / 256, 256, 0, stream>>>(yss, xg);

    // 8) out_proj + residual
    gemm_kernel<2><<<dim3(D_MODEL / 128, M_TOK / 128), 256, 0, stream>>>(
        xg + 2048, 2 * D_INNER, out_pw, D_INNER, out, D_MODEL, x, D_INNER);
}